// DifferentiableRAM_42382737277000
// MI455X (gfx1250) — compile-verified
//
#include <hip/hip_runtime.h>

typedef float v2f __attribute__((ext_vector_type(2)));
typedef float v8f __attribute__((ext_vector_type(8)));
typedef unsigned int u32x4 __attribute__((ext_vector_type(4)));
typedef int          i32x8 __attribute__((ext_vector_type(8)));
typedef int          i32x4 __attribute__((ext_vector_type(4)));

#define B_N   32
#define C_N   3
#define H_N   512
#define W_N   512
#define FAN_IN (C_N * H_N * W_N)      /* 786432 */
#define NCHUNK 96
#define CHUNK  (FAN_IN / NCHUNK)      /* 8192 */
#define NOUT   64

/* workspace layout in floats */
#define OFF_PART   0                                    /* 32*96*5 = 15360 */
#define OFF_PARAMS 16384                                /* 32*8           */
#define OFF_FY     32768                                /* 32*64*512      */
#define OFF_FXT    (OFF_FY  + B_N * NOUT * W_N)         /* 32*512*64      */
#define OFF_FYX    (OFF_FXT + B_N * W_N * NOUT)         /* 32*3*64*512    */

/* ---- Tensor Data Mover: 2D tile (f32) global -> LDS, with LDS row pad.
 * D# per CDNA5 ISA 8.3/8.4. pad_int_code: 5 -> every 64 DWORDs, 6 -> every
 * 128 DWORDs. pad_amt_code: 3 -> 4 DWORDs of padding. Groups 2/3 zero (2D).
 * This toolchain exposes the 6-arg builtin (extra int32x8 before cpol).   */
__device__ __forceinline__ void tdm_load_2d(const void* gsrc, unsigned lds_addr,
                                            unsigned tile_x, unsigned tile_y,
                                            unsigned tensor_x, unsigned tensor_y,
                                            unsigned stride0,
                                            unsigned pad_int_code, unsigned pad_amt_code)
{
    unsigned long long ga = (unsigned long long)(uintptr_t)gsrc;
    u32x4 g0;
    g0[0] = 1u;                                              /* count=1, user load  */
    g0[1] = lds_addr;                                        /* LDS byte address    */
    g0[2] = (unsigned)ga;                                    /* global_addr[31:0]   */
    g0[3] = ((unsigned)(ga >> 32) & 0x01FFFFFFu) | (2u << 30); /* ga[56:32], type=2 */
    i32x8 g1;
    g1[0] = (int)((2u << 16)                 /* data_size = 4B          */
                | (1u << 20)                 /* pad_enable              */
                | (pad_int_code << 22)
                | (pad_amt_code << 25));
    g1[1] = (int)((tensor_x & 0xFFFFu) << 16);               /* tensor_dim0[15:0]   */
    g1[2] = (int)((tensor_x >> 16) | ((tensor_y & 0xFFFFu) << 16));
    g1[3] = (int)((tensor_y >> 16) | (tile_x << 16));        /* tile_dim0           */
    g1[4] = (int)tile_y;                                     /* tile_dim1 (dim2=0)  */
    g1[5] = (int)stride0;                                    /* dim0 stride, units  */
    g1[6] = 0;
    g1[7] = 0;
    i32x4 z4 = {0, 0, 0, 0};
    i32x8 z8 = {0, 0, 0, 0, 0, 0, 0, 0};
    __builtin_amdgcn_tensor_load_to_lds(g0, g1, z4, z4, z8, 0);
}

/* ---------------- Kernel 1: locnet partial dot products -----------------
 * partial[b][chunk][j] = sum_{i in chunk} X[b][i] * W_loc[i][j]
 * Pure HBM stream (116 MB): float4-vectorized, prefetch-ahead, fixed-order
 * deterministic block tree reduce.                                        */
__global__ void k_locnet_partial(const float* __restrict__ X,
                                 const float* __restrict__ Wl,
                                 float* __restrict__ partial) {
    const int chunk = blockIdx.x;          // 0..95
    const int b     = blockIdx.y;          // 0..31
    const int t     = threadIdx.x;         // 0..255
    const float4* x4 = (const float4*)(X  + (size_t)b * FAN_IN + (size_t)chunk * CHUNK);
    const float4* w4 = (const float4*)(Wl + (size_t)chunk * CHUNK * 5);

    float acc[5] = {0.f, 0.f, 0.f, 0.f, 0.f};
    /* CHUNK/4 = 2048 float4 groups; thread t owns groups t, t+256, ...      */
    for (int i = t; i < CHUNK / 4; i += 256) {
        __builtin_prefetch((const void*)(x4 + i + 512), 0, 1);
        __builtin_prefetch((const void*)(w4 + (size_t)(i + 512) * 5), 0, 1);
        float4 x = x4[i];
        float4 q0 = w4[(size_t)i * 5 + 0];
        float4 q1 = w4[(size_t)i * 5 + 1];
        float4 q2 = w4[(size_t)i * 5 + 2];
        float4 q3 = w4[(size_t)i * 5 + 3];
        float4 q4 = w4[(size_t)i * 5 + 4];
        /* rows r=0..3 of W: offsets r*5+j over q0..q4 */
        acc[0] += x.x * q0.x;  acc[1] += x.x * q0.y;  acc[2] += x.x * q0.z;
        acc[3] += x.x * q0.w;  acc[4] += x.x * q1.x;
        acc[0] += x.y * q1.y;  acc[1] += x.y * q1.z;  acc[2] += x.y * q1.w;
        acc[3] += x.y * q2.x;  acc[4] += x.y * q2.y;
        acc[0] += x.z * q2.z;  acc[1] += x.z * q2.w;  acc[2] += x.z * q3.x;
        acc[3] += x.z * q3.y;  acc[4] += x.z * q3.z;
        acc[0] += x.w * q3.w;  acc[1] += x.w * q4.x;  acc[2] += x.w * q4.y;
        acc[3] += x.w * q4.z;  acc[4] += x.w * q4.w;
    }
    __shared__ float red[256];
    for (int j = 0; j < 5; ++j) {
        red[t] = acc[j];
        __syncthreads();
        for (int s = 128; s > 0; s >>= 1) {
            if (t < s) red[t] += red[t + s];
            __syncthreads();
        }
        if (t == 0) partial[((size_t)b * NCHUNK + chunk) * 5 + j] = red[0];
        __syncthreads();
    }
}

/* ---------------- Kernel 2: params + attention filter banks -------------
 * One block per batch; wave shuffle reductions (deterministic).
 * Fy row-major [b][m][h]; Fx stored transposed FxT[b][w][m].              */
__global__ void k_filters(const float* __restrict__ partial,
                          const float* __restrict__ b_loc,
                          float* __restrict__ Fy,
                          float* __restrict__ FxT,
                          float* __restrict__ params) {
    const int b = blockIdx.x;
    const int t = threadIdx.x;
    __shared__ float prm[5];

    if (t < 5) {
        float s = b_loc[t];
        for (int c = 0; c < NCHUNK; ++c)
            s += partial[((size_t)b * NCHUNK + c) * 5 + t];
        float v;
        if      (t == 0) v = 32.0f * (s + 1.0f);                  // gx
        else if (t == 1) v = 32.0f * (s + 1.0f);                  // gy
        else if (t == 2) v = 0.5f / expf(s);                      // 1/(2*sigma2)
        else if (t == 3) v = expf(s) * (511.0f / 63.0f);          // delta
        else             v = expf(s);                             // gamma
        prm[t] = v;
        params[(size_t)b * 8 + t] = v;
    }
    __syncthreads();

    const float gx = prm[0], gy = prm[1], inv2s = prm[2], delta = prm[3];
    const int lane = t & 31;
    const int wv   = t >> 5;

    for (int m = wv; m < NOUT; m += 8) {
        const float off = (float)m - 32.5f;      // i - N/2 - 0.5, N = 64
        const float mxv = gx + delta * off;
        const float myv = gy + delta * off;
        float ex[16], ey[16];
        float sx = 0.f, sy = 0.f;
#pragma unroll
        for (int j = 0; j < 16; ++j) {
            float a  = (float)(lane + 32 * j);
            float dx = a - mxv;  float vx = expf(-dx * dx * inv2s);
            float dy = a - myv;  float vy = expf(-dy * dy * inv2s);
            ex[j] = vx; sx += vx;
            ey[j] = vy; sy += vy;
        }
        for (int o = 16; o > 0; o >>= 1) {
            sx += __shfl_xor(sx, o, 32);
            sy += __shfl_xor(sy, o, 32);
        }
        const float rx = 1.0f / (sx + 1e-4f);
        const float ry = 1.0f / (sy + 1e-4f);
#pragma unroll
        for (int j = 0; j < 16; ++j) {
            int a = lane + 32 * j;
            FxT[((size_t)b * W_N + a) * NOUT + m] = ex[j] * rx;   // B operand of GEMM2
            Fy [((size_t)b * NOUT + m) * W_N + a] = ey[j] * ry;   // A operand of GEMM1
        }
    }
}

/* ---------------- Kernel 3: Fyx[b,c] = Fy[b] (64x512) @ X[b,c] (512x512)
 * f32 WMMA 16x16x4. Block = 8 waves, one 16-row m-tile x 128-col n-panel.
 * A tile (16x512, stride-516 pad) staged once cooperatively; B chunks
 * (32x128, padded to 132 by TDM) DMA'd by the Tensor Data Mover.          */
__global__ void k_gemm1(const float* __restrict__ X,
                        const float* __restrict__ Fy,
                        float* __restrict__ Fyx) {
    const int npan = blockIdx.x;           // 0..3  -> n0 = npan*128
    const int mt   = blockIdx.y;           // 0..3  -> m0 = mt*16
    const int bc   = blockIdx.z;           // 0..95
    const int b    = bc / C_N;
    const int c    = bc % C_N;
    const int t    = threadIdx.x;
    const int lane = t & 31;
    const int wv   = t >> 5;

    __shared__ float Al[16][516];          // Fy tile, padded: col reads hit distinct banks
    __shared__ float Bl[32][132];          // X chunk; TDM pads 4 DWORDs per 128

    const int m0  = mt * 16;
    const int n0p = npan * 128;
    const unsigned ldsB = (unsigned)(uintptr_t)&Bl[0][0];

    /* stage A: 16 x 512 = 8192 floats, 32 per thread, coalesced */
    for (int i = 0; i < 32; ++i) {
        int idx = t + i * 256;
        int r = idx >> 9, col = idx & 511;
        Al[r][col] = Fy[((size_t)b * NOUT + m0 + r) * W_N + col];
    }

    const int mrow  = lane & 15;
    const int khalf = (lane >> 4) * 2;
    v8f acc = {0.f, 0.f, 0.f, 0.f, 0.f, 0.f, 0.f, 0.f};

    for (int ck = 0; ck < 16; ++ck) {
        const int k0 = ck * 32;
        __syncthreads();                   /* previous chunk consumed */
        if (wv == 0) {
            const float* gsrc = X + (((size_t)b * C_N + c) * H_N + k0) * W_N + n0p;
            tdm_load_2d(gsrc, ldsB, /*tile*/128, 32, /*tensor*/512, 512,
                        /*stride*/512, /*pad every 128 dw*/6, /*by 4 dw*/3);
            __builtin_amdgcn_s_wait_tensorcnt((short)0);
        }
        __syncthreads();                   /* chunk visible to all waves */
#pragma unroll
        for (int kk = 0; kk < 32; kk += 4) {
            v2f a, bb;
            a.x  = Al[mrow][k0 + kk + khalf];
            a.y  = Al[mrow][k0 + kk + khalf + 1];
            bb.x = Bl[kk + khalf][wv * 16 + mrow];
            bb.y = Bl[kk + khalf + 1][wv * 16 + mrow];
            acc = __builtin_amdgcn_wmma_f32_16x16x4_f32(
                      false, a, false, bb, (short)0, acc, false, false);
        }
    }

    const int ccol = lane & 15;
#pragma unroll
    for (int r = 0; r < 8; ++r) {
        int row = r + 8 * (lane >> 4);
        Fyx[(((size_t)b * C_N + c) * NOUT + m0 + row) * W_N + n0p + wv * 16 + ccol] = acc[r];
    }
}

/* ---------------- Kernel 4: out[b,c] = gamma * Fyx[b,c] (64x512) @ FxT[b] (512x64)
 * One block per (b,c); 8 waves x 2 tiles cover the 4x4 grid of 16x16 tiles.
 * A and B 64x64 K-chunks DMA'd by the TDM (pad 4 DWORDs per 64 -> [64][68]). */
__global__ void k_gemm2(const float* __restrict__ Fyx,
                        const float* __restrict__ FxT,
                        const float* __restrict__ params,
                        float* __restrict__ out) {
    const int bc = blockIdx.x;             // 0..95
    const int b  = bc / C_N;
    const int c  = bc % C_N;
    const int t  = threadIdx.x;
    const int lane = t & 31;
    const int wv   = t >> 5;
    const float gamma = params[(size_t)b * 8 + 4];

    __shared__ float Al[64][68];
    __shared__ float Bl[64][68];
    const unsigned ldsA = (unsigned)(uintptr_t)&Al[0][0];
    const unsigned ldsB = (unsigned)(uintptr_t)&Bl[0][0];

    const int mrow  = lane & 15;
    const int khalf = (lane >> 4) * 2;

    const int rt0 = wv >> 2,       ct0 = wv & 3;        // tile wv
    const int rt1 = (wv + 8) >> 2, ct1 = (wv + 8) & 3;  // tile wv+8
    v8f acc0 = {0.f, 0.f, 0.f, 0.f, 0.f, 0.f, 0.f, 0.f};
    v8f acc1 = {0.f, 0.f, 0.f, 0.f, 0.f, 0.f, 0.f, 0.f};

    for (int ck = 0; ck < 8; ++ck) {
        const int k0 = ck * 64;
        __syncthreads();                   /* previous chunk consumed */
        if (wv == 0) {
            tdm_load_2d(Fyx + ((size_t)bc * NOUT) * W_N + k0, ldsA,
                        /*tile*/64, 64, /*tensor*/512, 64, /*stride*/512, 5, 3);
            tdm_load_2d(FxT + ((size_t)b * W_N + k0) * NOUT, ldsB,
                        /*tile*/64, 64, /*tensor*/64, 512, /*stride*/64, 5, 3);
            __builtin_amdgcn_s_wait_tensorcnt((short)0);
        }
        __syncthreads();                   /* chunks visible to all waves */
#pragma unroll
        for (int kk = 0; kk < 64; kk += 4) {
            v2f a, bb;
            a.x  = Al[rt0 * 16 + mrow][kk + khalf];
            a.y  = Al[rt0 * 16 + mrow][kk + khalf + 1];
            bb.x = Bl[kk + khalf][ct0 * 16 + mrow];
            bb.y = Bl[kk + khalf + 1][ct0 * 16 + mrow];
            acc0 = __builtin_amdgcn_wmma_f32_16x16x4_f32(
                       false, a, false, bb, (short)0, acc0, false, false);
            a.x  = Al[rt1 * 16 + mrow][kk + khalf];
            a.y  = Al[rt1 * 16 + mrow][kk + khalf + 1];
            bb.x = Bl[kk + khalf][ct1 * 16 + mrow];
            bb.y = Bl[kk + khalf + 1][ct1 * 16 + mrow];
            acc1 = __builtin_amdgcn_wmma_f32_16x16x4_f32(
                       false, a, false, bb, (short)0, acc1, false, false);
        }
    }

    const int ccol = lane & 15;
#pragma unroll
    for (int r = 0; r < 8; ++r) {
        int rr = r + 8 * (lane >> 4);
        out[(((size_t)b * C_N + c) * NOUT + rt0 * 16 + rr) * NOUT + ct0 * 16 + ccol] = gamma * acc0[r];
        out[(((size_t)b * C_N + c) * NOUT + rt1 * 16 + rr) * NOUT + ct1 * 16 + ccol] = gamma * acc1[r];
    }
}

extern "C" void kernel_launch(void* const* d_in, const int* in_sizes, int n_in,
                              void* d_out, int out_size, void* d_ws, size_t ws_size,
                              hipStream_t stream) {
    (void)in_sizes; (void)n_in; (void)out_size; (void)ws_size;
    const float* X     = (const float*)d_in[0];
    const float* W_loc = (const float*)d_in[1];
    const float* b_loc = (const float*)d_in[2];
    float* out = (float*)d_out;
    float* ws  = (float*)d_ws;

    float* partial = ws + OFF_PART;
    float* params  = ws + OFF_PARAMS;
    float* Fy      = ws + OFF_FY;
    float* FxT     = ws + OFF_FXT;
    float* Fyx     = ws + OFF_FYX;

    k_locnet_partial<<<dim3(NCHUNK, B_N), 256, 0, stream>>>(X, W_loc, partial);
    k_filters<<<dim3(B_N), 256, 0, stream>>>(partial, b_loc, Fy, FxT, params);
    k_gemm1<<<dim3(4, 4, B_N * C_N), 256, 0, stream>>>(X, Fy, Fyx);
    k_gemm2<<<dim3(B_N * C_N), 256, 0, stream>>>(Fyx, FxT, params, out);
}